// Detr3dCrossAtten_4466765988571
// MI455X (gfx1250) — compile-verified
//
#include <hip/hip_runtime.h>
#include <hip/hip_bf16.h>

typedef __attribute__((ext_vector_type(2))) float v2f;
typedef __attribute__((ext_vector_type(8))) float v8f;

// Problem constants (from setup_inputs)
constexpr int B_    = 2;
constexpr int NCAM  = 6;
constexpr int Q_    = 900;
constexpr int C_    = 256;
constexpr int H_    = 116;
constexpr int W_    = 200;
constexpr float EPS = 1e-05f;

// ---------------------------------------------------------------------------
// Kernel 1: fused projection + bilinear sampling + camera mean -> agg (B,Q,C)
// One block per (b,q). Threads 0..5 compute per-camera sample descriptors into
// LDS; all 256 threads then gather their channel across the 6 cameras.
// ---------------------------------------------------------------------------
__global__ void detr3d_sample_kernel(const float* __restrict__ ref_pts,   // (B,Q,3)
                                     const float* __restrict__ feats,     // (B,N,C,H,W)
                                     const float* __restrict__ l2i,       // (B,N,4,4)
                                     float* __restrict__ agg)             // (B*Q, C)
{
    __shared__ int   s_idx[NCAM][4];
    __shared__ float s_w[NCAM][4];

    const int bq = blockIdx.x;
    const int b  = bq / Q_;
    const int q  = bq - b * Q_;

    if (threadIdx.x < NCAM) {
        const int n = threadIdx.x;
        const float* M = l2i + ((size_t)b * NCAM + n) * 16;
        const float* r = ref_pts + ((size_t)b * Q_ + q) * 3;
        const float px = r[0], py = r[1], pz = r[2];

        const float p0 = M[0] * px + M[1] * py + M[2]  * pz + M[3];
        const float p1 = M[4] * px + M[5] * py + M[6]  * pz + M[7];
        const float z  = M[8] * px + M[9] * py + M[10] * pz + M[11];

        const float invd = 1.0f / (fabsf(z) + EPS);
        const float gx = (p0 * invd) / (float)(W_ - 1) * 2.0f - 1.0f;
        const float gy = (p1 * invd) / (float)(H_ - 1) * 2.0f - 1.0f;

        const bool valid = (z > EPS) && (fmaxf(fabsf(gx), fabsf(gy)) <= 1.0f);

        float x = ((gx + 1.0f) * (float)W_ - 1.0f) * 0.5f;
        float y = ((gy + 1.0f) * (float)H_ - 1.0f) * 0.5f;
        if (!valid) { x = 0.0f; y = 0.0f; }

        const float x0f = floorf(x), y0f = floorf(y);
        const float wx = x - x0f, wy = y - y0f;
        const int ix0 = (int)x0f, iy0 = (int)y0f;
        const float vmask = valid ? 1.0f : 0.0f;

        const float cw[4] = { (1.0f - wx) * (1.0f - wy), wx * (1.0f - wy),
                              (1.0f - wx) * wy,          wx * wy };
        const int dx[4] = { 0, 1, 0, 1 };
        const int dy[4] = { 0, 0, 1, 1 };
#pragma unroll
        for (int k = 0; k < 4; ++k) {
            const int ix = ix0 + dx[k];
            const int iy = iy0 + dy[k];
            const bool in = (ix >= 0) && (ix < W_) && (iy >= 0) && (iy < H_);
            const int cx = min(max(ix, 0), W_ - 1);
            const int cy = min(max(iy, 0), H_ - 1);
            s_idx[n][k] = cy * W_ + cx;
            s_w[n][k]   = cw[k] * (in ? 1.0f : 0.0f) * vmask;
        }
    }
    __syncthreads();

    const int c = threadIdx.x;                  // one channel per thread
    const size_t HW = (size_t)H_ * W_;
    float sum = 0.0f;
#pragma unroll
    for (int n = 0; n < NCAM; ++n) {
        const float* base = feats + (((size_t)b * NCAM + n) * C_ + c) * HW;
        sum += s_w[n][0] * base[s_idx[n][0]]
             + s_w[n][1] * base[s_idx[n][1]]
             + s_w[n][2] * base[s_idx[n][2]]
             + s_w[n][3] * base[s_idx[n][3]];
    }
    agg[(size_t)bq * C_ + c] = sum * (1.0f / (float)NCAM);
}

// ---------------------------------------------------------------------------
// Kernel 2: out = agg @ W_out^T + b_out via V_WMMA_F32_16X16X4_F32 (f32 WMMA).
// M = B*Q = 1800 (edge tile handled by clamped loads + guarded stores),
// N = K = 256. Each wave computes a 16x64 strip: 4 accumulators sharing one
// A fragment.
// A frag (16x4 f32): lane = M row; lanes 0-15 hold K=k0+0,k0+1 in v[0],v[1];
//                    lanes 16-31 hold K=k0+2,k0+3.
// B frag (4x16):     lane = N col, same K-half split -> contiguous W[j][k..k+1].
//
// Edge handling: the A row index is CLAMPED (not masked) — an out-of-range
// lane just re-reads the last valid row. Its garbage lands only in D rows we
// never store, so the K loop stays branch-free with full EXEC (required for
// WMMA) and no per-iteration exec save/restore.
// ---------------------------------------------------------------------------
__global__ void detr3d_out_gemm_kernel(const float* __restrict__ A,     // (M, 256) agg
                                       const float* __restrict__ Wout,  // (256, 256) W[j][k]
                                       const float* __restrict__ bias,  // (256)
                                       float* __restrict__ out,         // (M, 256)
                                       int Mrows)
{
    const int lane    = threadIdx.x & 31;
    const int wave    = threadIdx.x >> 5;
    const int waveId  = blockIdx.x * (blockDim.x >> 5) + wave;
    const int mTile   = waveId >> 2;         // 16-row tile index
    const int nQuad   = waveId & 3;          // which 64-column strip
    if (mTile * 16 >= Mrows) return;         // whole-wave exit -> EXEC all 1s below

    const int halfSel = lane >> 4;           // K-half select within fragment
    const int l       = lane & 15;
    const int m       = min(mTile * 16 + l, Mrows - 1);  // clamped A row
    const int n0      = nQuad * 64;

    const float* aRow = A + (size_t)m * C_ + halfSel * 2;
    const float* bRow0 = Wout + (size_t)(n0 +  0 + l) * C_ + halfSel * 2;
    const float* bRow1 = bRow0 + 16 * C_;
    const float* bRow2 = bRow0 + 32 * C_;
    const float* bRow3 = bRow0 + 48 * C_;

    v8f acc0 = {}, acc1 = {}, acc2 = {}, acc3 = {};

    for (int k0 = 0; k0 < C_; k0 += 4) {
        const v2f a  = *(const v2f*)(aRow  + k0);
        const v2f b0 = *(const v2f*)(bRow0 + k0);
        const v2f b1 = *(const v2f*)(bRow1 + k0);
        const v2f b2 = *(const v2f*)(bRow2 + k0);
        const v2f b3 = *(const v2f*)(bRow3 + k0);

        acc0 = __builtin_amdgcn_wmma_f32_16x16x4_f32(false, a, false, b0, (short)0, acc0, false, false);
        acc1 = __builtin_amdgcn_wmma_f32_16x16x4_f32(false, a, false, b1, (short)0, acc1, false, false);
        acc2 = __builtin_amdgcn_wmma_f32_16x16x4_f32(false, a, false, b2, (short)0, acc2, false, false);
        acc3 = __builtin_amdgcn_wmma_f32_16x16x4_f32(false, a, false, b3, (short)0, acc3, false, false);
    }

    // C/D layout: VGPR r, lanes 0-15 -> (M = 16*mTile + r,      N = lane)
    //                     lanes 16-31 -> (M = 16*mTile + 8 + r, N = lane-16)
    const int col     = l;
    const int rowBase = mTile * 16 + halfSel * 8;
#pragma unroll
    for (int r = 0; r < 8; ++r) {
        const int row = rowBase + r;
        if (row < Mrows) {
            float* o = out + (size_t)row * C_ + n0 + col;
            o[0]  = acc0[r] + bias[n0 +  0 + col];
            o[16] = acc1[r] + bias[n0 + 16 + col];
            o[32] = acc2[r] + bias[n0 + 32 + col];
            o[48] = acc3[r] + bias[n0 + 48 + col];
        }
    }
}

extern "C" void kernel_launch(void* const* d_in, const int* in_sizes, int n_in,
                              void* d_out, int out_size, void* d_ws, size_t ws_size,
                              hipStream_t stream) {
    // Inputs (setup_inputs order): query, reference_points, image_features,
    // lidar2img, W_out, b_out. `query` is unused by the reference math.
    const float* ref_pts = (const float*)d_in[1];
    const float* feats   = (const float*)d_in[2];
    const float* l2i     = (const float*)d_in[3];
    const float* Wout    = (const float*)d_in[4];
    const float* bias    = (const float*)d_in[5];
    float*       out     = (float*)d_out;

    float* agg = (float*)d_ws;               // (B*Q, C) f32 scratch, 1.8 MB

    // Kernel 1: one block per (b,q), one channel per thread.
    detr3d_sample_kernel<<<B_ * Q_, C_, 0, stream>>>(ref_pts, feats, l2i, agg);

    // Kernel 2: WMMA GEMM. M=1800 -> 113 m-tiles x 4 n-strips = 452 waves.
    const int Mrows   = B_ * Q_;
    const int mTiles  = (Mrows + 15) / 16;
    const int waves   = mTiles * 4;
    const int wavesPerBlock = 8;             // 256 threads
    const int blocks  = (waves + wavesPerBlock - 1) / wavesPerBlock;
    detr3d_out_gemm_kernel<<<blocks, wavesPerBlock * 32, 0, stream>>>(
        agg, Wout, bias, out, Mrows);
}